// GraphConvolutionalNetwork_13073880449170
// MI455X (gfx1250) — compile-verified
//
#include <hip/hip_runtime.h>

#define NN 50000
#define DIM 64
#define KDIM 256        // 4 * DIM concatenated features
#define WT_STRIDE 260   // LDS row stride (words): ≡4 mod 64 -> conflict-free ds_load_b64
#define WT_BYTES (DIM * WT_STRIDE * 4)   // 66,560 B dynamic LDS

typedef __attribute__((ext_vector_type(2))) float v2f;
typedef __attribute__((ext_vector_type(8))) float v8f;

// ---------------------------------------------------------------- zero fill
__global__ void zero_kernel(float* __restrict__ p, size_t n) {
  size_t i = (size_t)blockIdx.x * blockDim.x + threadIdx.x;
  const size_t stride = (size_t)gridDim.x * blockDim.x;
  for (; i < n; i += stride) p[i] = 0.0f;
}

// ---------------------------------------------------------------- degree counts
__global__ void count_kernel(const int* __restrict__ row, const int* __restrict__ col,
                             float* __restrict__ cnt_row, float* __restrict__ cnt_col, int E) {
  const int e = blockIdx.x * blockDim.x + threadIdx.x;
  if (e >= E) return;
  atomicAdd(&cnt_row[row[e]], 1.0f);
  atomicAdd(&cnt_col[col[e]], 1.0f);
}

// ---------------------------------------------------------------- edge scatter
// 64 threads per edge (j = column). acc is [N][256] laid out as
// [ out(0:64) | opinion(64:128) | inn(128:192) | inn_opinion(192:256) ].
__global__ __launch_bounds__(256) void edge_scatter_kernel(
    const float* __restrict__ x, const int* __restrict__ row, const int* __restrict__ col,
    const float* __restrict__ el, const float* __restrict__ tw,
    float* __restrict__ acc, int E) {
  const long long t = (long long)blockIdx.x * blockDim.x + threadIdx.x;
  const int e = (int)(t >> 6);
  const int j = (int)(t & 63);
  if (e >= E) return;
  const int r = row[e];
  const int c = col[e];
  const float e0 = el[2 * e + 0];
  const float e1 = el[2 * e + 1];
  const float eltj = fmaf(e0, tw[j], e1 * tw[64 + j]);
  const float xc = x[(size_t)c * DIM + j];   // -> out   (scatter by row)
  const float xr = x[(size_t)r * DIM + j];   // -> inn   (scatter by col)
  atomicAdd(acc + (size_t)r * KDIM + j,        xc);
  atomicAdd(acc + (size_t)r * KDIM + 64 + j,  eltj);
  atomicAdd(acc + (size_t)c * KDIM + 128 + j,  xr);
  atomicAdd(acc + (size_t)c * KDIM + 192 + j, eltj);
}

// ---------------------------------------------------------------- WMMA GEMM
// One wave computes a 16-node x 64-col tile of relu( (acc ./ deg) @ w + b ).
// Normalization is deferred to the epilogue (GEMM is linear):
//   D_row = sum_{k<128} h.w   (out|opinion block, scaled by 1/max(cnt_row,1))
//   D_col = sum_{k>=128} h.w  (inn|inn_opinion block, scaled by 1/max(cnt_col,1))
// Inner loop body = 1 global_load_b64 (A) + 4 ds_load_b64 (B) + 4 v_wmma.
__global__ __launch_bounds__(256) void gemm_bias_relu_kernel(
    const float* __restrict__ acc,      // [N][256]
    const float* __restrict__ w,        // [256][64]
    const float* __restrict__ bias,     // [64]
    const float* __restrict__ cnt_row,
    const float* __restrict__ cnt_col,
    float* __restrict__ out,            // [N][64]
    int nnodes) {
  extern __shared__ float wT[];         // [DIM][WT_STRIDE] transposed weights

  // cooperative transposed stage of w (coalesced global reads)
  for (int i = threadIdx.x; i < KDIM * DIM; i += 256) {
    const int k = i >> 6;               // 0..255
    const int n = i & 63;
    wT[n * WT_STRIDE + k] = w[i];
  }
  __syncthreads();

  const int lane  = (int)(threadIdx.x & 31);
  const int wave  = blockIdx.x * 8 + (int)(threadIdx.x >> 5);
  const int mbase = wave * 16;
  if (mbase >= nnodes) return;          // wave-uniform; EXEC stays all-1s

  const int mloc = lane & 15;
  const int hi   = lane >> 4;           // half-wave selector
  const float* __restrict__ hrow = acc + (size_t)(mbase + mloc) * KDIM;
  const float* __restrict__ bT0 = wT + (size_t)(mloc)      * WT_STRIDE;
  const float* __restrict__ bT1 = wT + (size_t)(mloc + 16) * WT_STRIDE;
  const float* __restrict__ bT2 = wT + (size_t)(mloc + 32) * WT_STRIDE;
  const float* __restrict__ bT3 = wT + (size_t)(mloc + 48) * WT_STRIDE;

  v8f cA0 = {}, cA1 = {}, cA2 = {}, cA3 = {};   // k in [0,128)
  v8f cB0 = {}, cB1 = {}, cB2 = {}, cB3 = {};   // k in [128,256)

  // A 16x4 layout: lanes 0-15 hold K=k0,k0+1; lanes 16-31 hold K=k0+2,k0+3.
  // B 4x16 layout: symmetric K split across half-waves, N = lane&15 (+tile).
#pragma unroll 4
  for (int k0 = 0; k0 < 128; k0 += 4) {
    const int ka = k0 + hi * 2;
    const v2f a  = *reinterpret_cast<const v2f*>(hrow + ka);
    const v2f b0 = *reinterpret_cast<const v2f*>(bT0 + ka);
    const v2f b1 = *reinterpret_cast<const v2f*>(bT1 + ka);
    const v2f b2 = *reinterpret_cast<const v2f*>(bT2 + ka);
    const v2f b3 = *reinterpret_cast<const v2f*>(bT3 + ka);
    cA0 = __builtin_amdgcn_wmma_f32_16x16x4_f32(false, a, false, b0, (short)0, cA0, false, false);
    cA1 = __builtin_amdgcn_wmma_f32_16x16x4_f32(false, a, false, b1, (short)0, cA1, false, false);
    cA2 = __builtin_amdgcn_wmma_f32_16x16x4_f32(false, a, false, b2, (short)0, cA2, false, false);
    cA3 = __builtin_amdgcn_wmma_f32_16x16x4_f32(false, a, false, b3, (short)0, cA3, false, false);
  }
#pragma unroll 4
  for (int k0 = 128; k0 < 256; k0 += 4) {
    const int ka = k0 + hi * 2;
    const v2f a  = *reinterpret_cast<const v2f*>(hrow + ka);
    const v2f b0 = *reinterpret_cast<const v2f*>(bT0 + ka);
    const v2f b1 = *reinterpret_cast<const v2f*>(bT1 + ka);
    const v2f b2 = *reinterpret_cast<const v2f*>(bT2 + ka);
    const v2f b3 = *reinterpret_cast<const v2f*>(bT3 + ka);
    cB0 = __builtin_amdgcn_wmma_f32_16x16x4_f32(false, a, false, b0, (short)0, cB0, false, false);
    cB1 = __builtin_amdgcn_wmma_f32_16x16x4_f32(false, a, false, b1, (short)0, cB1, false, false);
    cB2 = __builtin_amdgcn_wmma_f32_16x16x4_f32(false, a, false, b2, (short)0, cB2, false, false);
    cB3 = __builtin_amdgcn_wmma_f32_16x16x4_f32(false, a, false, b3, (short)0, cB3, false, false);
  }

  // D layout: VGPR g -> M = mbase + 8*hi + g, N = lane&15 (+ tile offset).
  const float bv0 = bias[mloc];
  const float bv1 = bias[mloc + 16];
  const float bv2 = bias[mloc + 32];
  const float bv3 = bias[mloc + 48];
  const int rowoff = hi * 8;
#pragma unroll
  for (int g = 0; g < 8; ++g) {
    const int mg = mbase + rowoff + g;
    const float ir = 1.0f / fmaxf(cnt_row[mg], 1.0f);   // broadcast load
    const float ic = 1.0f / fmaxf(cnt_col[mg], 1.0f);
    const size_t mm = (size_t)mg * DIM;
    out[mm + mloc     ] = fmaxf(fmaf(cA0[g], ir, fmaf(cB0[g], ic, bv0)), 0.0f);
    out[mm + mloc + 16] = fmaxf(fmaf(cA1[g], ir, fmaf(cB1[g], ic, bv1)), 0.0f);
    out[mm + mloc + 32] = fmaxf(fmaf(cA2[g], ir, fmaf(cB2[g], ic, bv2)), 0.0f);
    out[mm + mloc + 48] = fmaxf(fmaf(cA3[g], ir, fmaf(cB3[g], ic, bv3)), 0.0f);
  }
}

// ---------------------------------------------------------------- launch
extern "C" void kernel_launch(void* const* d_in, const int* in_sizes, int n_in,
                              void* d_out, int out_size, void* d_ws, size_t ws_size,
                              hipStream_t stream) {
  const float* X   = (const float*)d_in[0];
  const int*   ei  = (const int*)d_in[1];
  const float* el  = (const float*)d_in[2];
  const float* w1  = (const float*)d_in[3];
  const float* tw1 = (const float*)d_in[4];
  const float* b1  = (const float*)d_in[5];
  const float* w2  = (const float*)d_in[6];
  const float* tw2 = (const float*)d_in[7];
  const float* b2  = (const float*)d_in[8];

  const int E = in_sizes[1] / 2;
  const int N = NN;
  const int* row = ei;
  const int* col = ei + E;

  // workspace layout
  float* acc     = (float*)d_ws;                 // N*256
  float* cnt_row = acc + (size_t)N * KDIM;       // N
  float* cnt_col = cnt_row + N;                  // N
  float* h1      = cnt_col + N;                  // N*64

  const int ZB = 2048;
  const size_t accN = (size_t)N * KDIM;

  // degree counts (shared by both layers)
  zero_kernel<<<ZB, 256, 0, stream>>>(cnt_row, (size_t)2 * N);
  count_kernel<<<(E + 255) / 256, 256, 0, stream>>>(row, col, cnt_row, cnt_col, E);

  const int scatter_blocks = (int)(((long long)E * 64 + 255) / 256);
  const int gemm_blocks = (N / 16 + 7) / 8;

  // ---- layer 1: X -> h1
  zero_kernel<<<ZB, 256, 0, stream>>>(acc, accN);
  edge_scatter_kernel<<<scatter_blocks, 256, 0, stream>>>(X, row, col, el, tw1, acc, E);
  gemm_bias_relu_kernel<<<gemm_blocks, 256, WT_BYTES, stream>>>(acc, w1, b1, cnt_row, cnt_col,
                                                                h1, N);

  // ---- layer 2: h1 -> out
  zero_kernel<<<ZB, 256, 0, stream>>>(acc, accN);
  edge_scatter_kernel<<<scatter_blocks, 256, 0, stream>>>(h1, row, col, el, tw2, acc, E);
  gemm_bias_relu_kernel<<<gemm_blocks, 256, WT_BYTES, stream>>>(acc, w2, b2, cnt_row, cnt_col,
                                                                (float*)d_out, N);
}